// GATNet_62491774157296
// MI455X (gfx1250) — compile-verified
//
#include <hip/hip_runtime.h>

// ---------------------------------------------------------------------------
// GATNet forward for MI455X (gfx1250).  fp32 everywhere (matches reference);
// dense GEMMs use V_WMMA_F32_16X16X4_F32; edge softmax/scatter via L2 atomics.
// ---------------------------------------------------------------------------

typedef __attribute__((ext_vector_type(2))) float v2f;
typedef __attribute__((ext_vector_type(8))) float v8f;

#define NEG_SLOPE 0.2f

// ---- order-preserving float <-> uint key (for atomicMax on floats) --------
__device__ __forceinline__ unsigned f2key(float f) {
  unsigned u = __float_as_uint(f);
  return (u & 0x80000000u) ? ~u : (u | 0x80000000u);
}
__device__ __forceinline__ float key2f(unsigned k) {
  return (k & 0x80000000u) ? __uint_as_float(k & 0x7fffffffu)
                           : __uint_as_float(~k);
}

// ---------------------------------------------------------------------------
// WMMA fp32 GEMM:  C[M x 64] = A[M x K] * B[K x 64]
// One wave handles a 16-row tile with 4 accumulators (16x16 each) across N=64.
// A frag (16x4 fp32):  lane<16 -> K={k,k+1}; lane>=16 -> K={k+2,k+3}
// B frag (4x16 fp32):  lane%16 = N; lane-half selects K pair; vgpr selects in pair
// C/D (16x16 fp32):    vgpr r: lanes 0-15 -> M=r, lanes 16-31 -> M=r+8; N=lane%16
// ---------------------------------------------------------------------------
template <int K>
__global__ void gemm_wmma(const float* __restrict__ A, const float* __restrict__ B,
                          float* __restrict__ C, int M) {
  const int lane = threadIdx.x & 31;
  const int wave = threadIdx.x >> 5;
  const int tile = blockIdx.x * (blockDim.x >> 5) + wave;
  const int row0 = tile << 4;
  if (row0 >= M) return;                 // wave-uniform: EXEC stays all-ones
  const int hs = lane >> 4;              // lane-half select
  const int lm = lane & 15;

  v8f c0 = {}; v8f c1 = {}; v8f c2 = {}; v8f c3 = {};
  const float* Ar = A + (size_t)(row0 + lm) * K;

#pragma unroll
  for (int k = 0; k < K; k += 4) {
    const int ka = k + hs * 2;
    v2f a;
    a.x = Ar[ka];
    a.y = Ar[ka + 1];
    const float* B0 = B + (size_t)ka * 64;   // row K=ka
    const float* B1 = B0 + 64;               // row K=ka+1
    v2f b0, b1, b2, b3;
    b0.x = B0[lm];       b0.y = B1[lm];
    b1.x = B0[16 + lm];  b1.y = B1[16 + lm];
    b2.x = B0[32 + lm];  b2.y = B1[32 + lm];
    b3.x = B0[48 + lm];  b3.y = B1[48 + lm];
    c0 = __builtin_amdgcn_wmma_f32_16x16x4_f32(false, a, false, b0, (short)0, c0, false, false);
    c1 = __builtin_amdgcn_wmma_f32_16x16x4_f32(false, a, false, b1, (short)0, c1, false, false);
    c2 = __builtin_amdgcn_wmma_f32_16x16x4_f32(false, a, false, b2, (short)0, c2, false, false);
    c3 = __builtin_amdgcn_wmma_f32_16x16x4_f32(false, a, false, b3, (short)0, c3, false, false);
  }

#pragma unroll
  for (int r = 0; r < 8; ++r) {
    const int row = row0 + r + hs * 8;
    float* Cr = C + (size_t)row * 64 + lm;
    Cr[0]  = c0[r];
    Cr[16] = c1[r];
    Cr[32] = c2[r];
    Cr[48] = c3[r];
  }
}

// ---- per-(node, head) attention coefficients ------------------------------
__global__ void attn_coef(const float* __restrict__ h, const float* __restrict__ att_s,
                          const float* __restrict__ att_d, float* __restrict__ a_src,
                          float* __restrict__ a_dst, int N, int H, int C) {
  int t = blockIdx.x * blockDim.x + threadIdx.x;
  if (t >= N * H) return;
  int n = t / H, hh = t - n * H;
  const float* hp = h + (size_t)n * H * C + (size_t)hh * C;
  const float* sp = att_s + hh * C;
  const float* dp = att_d + hh * C;
  float s = 0.f, d = 0.f;
  for (int c = 0; c < C; ++c) { float v = hp[c]; s += v * sp[c]; d += v * dp[c]; }
  a_src[t] = s;
  a_dst[t] = d;
}

// ---- fills ----------------------------------------------------------------
__global__ void fillf_k(float* __restrict__ p, float v, long n) {
  long t = (long)blockIdx.x * blockDim.x + threadIdx.x;
  if (t < n) p[t] = v;
}
__global__ void fillkey_k(unsigned* __restrict__ p, float v, long n) {
  long t = (long)blockIdx.x * blockDim.x + threadIdx.x;
  if (t < n) p[t] = f2key(v);
}

// ---- edge pass 1: segment max of leaky_relu(a_src[s]+a_dst[d]) ------------
__global__ void edge_max_k(const int* __restrict__ ei, const float* __restrict__ a_src,
                           const float* __restrict__ a_dst, unsigned* __restrict__ dmax,
                           int E, int N, int H) {
  long t = (long)blockIdx.x * blockDim.x + threadIdx.x;
  long tot = (long)(E + N) * H;
  if (t >= tot) return;
  int e = (int)(t / H), hh = (int)(t - (long)e * H);
  int s, d;
  if (e < E) { s = ei[e]; d = ei[E + e]; } else { s = e - E; d = s; }
  float x = a_src[(size_t)s * H + hh] + a_dst[(size_t)d * H + hh];
  x = x >= 0.f ? x : NEG_SLOPE * x;
  atomicMax(&dmax[(size_t)d * H + hh], f2key(x));
}

// ---- edge pass 2: segment sum of exp(e - max) -----------------------------
__global__ void edge_sum_k(const int* __restrict__ ei, const float* __restrict__ a_src,
                           const float* __restrict__ a_dst, const unsigned* __restrict__ dmax,
                           float* __restrict__ denom, int E, int N, int H) {
  long t = (long)blockIdx.x * blockDim.x + threadIdx.x;
  long tot = (long)(E + N) * H;
  if (t >= tot) return;
  int e = (int)(t / H), hh = (int)(t - (long)e * H);
  int s, d;
  if (e < E) { s = ei[e]; d = ei[E + e]; } else { s = e - E; d = s; }
  float x = a_src[(size_t)s * H + hh] + a_dst[(size_t)d * H + hh];
  x = x >= 0.f ? x : NEG_SLOPE * x;
  size_t di = (size_t)d * H + hh;
  atomicAdd(&denom[di], __expf(x - key2f(dmax[di])));
}

// ---- edge pass 3: out[dst] += h[src] * alpha ------------------------------
__global__ void edge_scatter_k(const int* __restrict__ ei, const float* __restrict__ h,
                               const float* __restrict__ a_src, const float* __restrict__ a_dst,
                               const unsigned* __restrict__ dmax, const float* __restrict__ denom,
                               float* __restrict__ out, int E, int N, int H, int C) {
  long t = (long)blockIdx.x * blockDim.x + threadIdx.x;
  long tot = (long)(E + N) * H;
  if (t >= tot) return;
  int e = (int)(t / H), hh = (int)(t - (long)e * H);
  int s, d;
  if (e < E) { s = ei[e]; d = ei[E + e]; } else { s = e - E; d = s; }
  float x = a_src[(size_t)s * H + hh] + a_dst[(size_t)d * H + hh];
  x = x >= 0.f ? x : NEG_SLOPE * x;
  size_t di = (size_t)d * H + hh;
  float ex = __expf(x - key2f(dmax[di]));
  float alpha = ex / fmaxf(denom[di], 1e-16f);
  const float* hp = h + (size_t)s * H * C + (size_t)hh * C;
  float* op = out + (size_t)d * H * C + (size_t)hh * C;
  for (int c = 0; c < C; ++c) atomicAdd(&op[c], hp[c] * alpha);
}

// ---- bias (+ optional ELU), in place --------------------------------------
__global__ void bias_act_k(float* __restrict__ y, const float* __restrict__ b,
                           long n, int F, int do_elu) {
  long t = (long)blockIdx.x * blockDim.x + threadIdx.x;
  if (t >= n) return;
  int f = (int)(t % F);
  float v = y[t] + b[f];
  if (do_elu) v = v > 0.f ? v : (__expf(v) - 1.f);
  y[t] = v;
}

// ---- global mean pool (accumulate) ----------------------------------------
__global__ void pool_k(const float* __restrict__ h, const int* __restrict__ batch,
                       float* __restrict__ sums, float* __restrict__ cnt, int N, int F) {
  int n = blockIdx.x * blockDim.x + threadIdx.x;
  if (n >= N) return;
  int g = batch[n];
  const float* hp = h + (size_t)n * F;
  float* sp = sums + (size_t)g * F;
  for (int c = 0; c < F; ++c) atomicAdd(&sp[c], hp[c]);
  atomicAdd(&cnt[g], 1.f);
}

// ---- final linear: out[g,k] = (sums[g,:]/cnt[g]) . lin_w[:,k] + lin_b[k] --
__global__ void final_k(const float* __restrict__ sums, const float* __restrict__ cnt,
                        const float* __restrict__ w, const float* __restrict__ b,
                        float* __restrict__ out, int G, int F, int K) {
  int t = blockIdx.x * blockDim.x + threadIdx.x;
  if (t >= G * K) return;
  int g = t / K, k = t - g * K;
  float inv = 1.f / fmaxf(cnt[g], 1.f);
  const float* sp = sums + (size_t)g * F;
  float acc = b[k];
  for (int c = 0; c < F; ++c) acc += sp[c] * inv * w[c * K + k];
  out[t] = acc;
}

// ---------------------------------------------------------------------------
extern "C" void kernel_launch(void* const* d_in, const int* in_sizes, int n_in,
                              void* d_out, int out_size, void* d_ws, size_t ws_size,
                              hipStream_t stream) {
  const float* x   = (const float*)d_in[0];
  const int*   ei  = (const int*)d_in[1];
  const int*   bat = (const int*)d_in[2];
  const float* W1  = (const float*)d_in[3];
  const float* as1 = (const float*)d_in[4];
  const float* ad1 = (const float*)d_in[5];
  const float* b1  = (const float*)d_in[6];
  const float* W2  = (const float*)d_in[7];
  const float* as2 = (const float*)d_in[8];
  const float* ad2 = (const float*)d_in[9];
  const float* b2  = (const float*)d_in[10];
  const float* lw  = (const float*)d_in[11];
  const float* lb  = (const float*)d_in[12];
  float* out = (float*)d_out;

  const int N   = in_sizes[0] / 128;   // 50000
  const int E   = in_sizes[1] / 2;     // 800000
  const int CLS = in_sizes[12];        // 10
  const int G   = out_size / CLS;      // 256
  (void)n_in; (void)ws_size;

  // workspace layout (256B-aligned slabs)
  char* ws = (char*)d_ws;
  size_t off = 0;
  auto wsalloc = [&](size_t bytes) -> void* {
    void* p = ws + off;
    off += (bytes + 255) & ~(size_t)255;
    return p;
  };
  float*    h     = (float*)   wsalloc((size_t)N * 64 * 4);  // h1 then h3
  float*    agg   = (float*)   wsalloc((size_t)N * 64 * 4);  // agg1/act1, then agg2
  float*    asrc  = (float*)   wsalloc((size_t)N * 8 * 4);
  float*    adst  = (float*)   wsalloc((size_t)N * 8 * 4);
  unsigned* dmax  = (unsigned*)wsalloc((size_t)N * 8 * 4);
  float*    denom = (float*)   wsalloc((size_t)N * 8 * 4);
  float*    sums  = (float*)   wsalloc((size_t)G * 64 * 4);
  float*    cnt   = (float*)   wsalloc((size_t)G * 4);

  const int TB = 256;
  auto cdiv = [](long a, long b) -> unsigned { return (unsigned)((a + b - 1) / b); };

  const int  tiles = (N + 15) / 16;
  const long tot1  = (long)(E + N) * 8;   // layer-1 edge-head work items
  const long tot2  = (long)(E + N) * 1;   // layer-2 edge work items

  // ======================= Layer 1 (128 -> 8x8, concat) ====================
  gemm_wmma<128><<<cdiv(tiles, 4), 128, 0, stream>>>(x, W1, h, N);
  attn_coef<<<cdiv((long)N * 8, TB), TB, 0, stream>>>(h, as1, ad1, asrc, adst, N, 8, 8);
  fillkey_k<<<cdiv((long)N * 8, TB), TB, 0, stream>>>(dmax, -3.0e38f, (long)N * 8);
  fillf_k  <<<cdiv((long)N * 8, TB), TB, 0, stream>>>(denom, 0.f, (long)N * 8);
  fillf_k  <<<cdiv((long)N * 64, TB), TB, 0, stream>>>(agg, 0.f, (long)N * 64);
  edge_max_k    <<<cdiv(tot1, TB), TB, 0, stream>>>(ei, asrc, adst, dmax, E, N, 8);
  edge_sum_k    <<<cdiv(tot1, TB), TB, 0, stream>>>(ei, asrc, adst, dmax, denom, E, N, 8);
  edge_scatter_k<<<cdiv(tot1, TB), TB, 0, stream>>>(ei, h, asrc, adst, dmax, denom, agg, E, N, 8, 8);
  bias_act_k<<<cdiv((long)N * 64, TB), TB, 0, stream>>>(agg, b1, (long)N * 64, 64, 1);

  // ======================= Layer 2 (64 -> 1x64, mean) ======================
  gemm_wmma<64><<<cdiv(tiles, 4), 128, 0, stream>>>(agg, W2, h, N);
  attn_coef<<<cdiv((long)N, TB), TB, 0, stream>>>(h, as2, ad2, asrc, adst, N, 1, 64);
  fillkey_k<<<cdiv((long)N, TB), TB, 0, stream>>>(dmax, -3.0e38f, (long)N);
  fillf_k  <<<cdiv((long)N, TB), TB, 0, stream>>>(denom, 0.f, (long)N);
  fillf_k  <<<cdiv((long)N * 64, TB), TB, 0, stream>>>(agg, 0.f, (long)N * 64);
  edge_max_k    <<<cdiv(tot2, TB), TB, 0, stream>>>(ei, asrc, adst, dmax, E, N, 1);
  edge_sum_k    <<<cdiv(tot2, TB), TB, 0, stream>>>(ei, asrc, adst, dmax, denom, E, N, 1);
  edge_scatter_k<<<cdiv(tot2, TB), TB, 0, stream>>>(ei, h, asrc, adst, dmax, denom, agg, E, N, 1, 64);
  bias_act_k<<<cdiv((long)N * 64, TB), TB, 0, stream>>>(agg, b2, (long)N * 64, 64, 0);

  // ======================= Pool + classifier ===============================
  fillf_k<<<cdiv((long)G * 64, TB), TB, 0, stream>>>(sums, 0.f, (long)G * 64);
  fillf_k<<<cdiv((long)G, TB), TB, 0, stream>>>(cnt, 0.f, (long)G);
  pool_k<<<cdiv((long)N, TB), TB, 0, stream>>>(agg, bat, sums, cnt, N, 64);
  final_k<<<cdiv((long)G * CLS, TB), TB, 0, stream>>>(sums, cnt, lw, lb, out, G, 64, CLS);
}